// DepatchSampling_2216203124766
// MI455X (gfx1250) — compile-verified
//
#include <hip/hip_runtime.h>
#include <math.h>

typedef __attribute__((ext_vector_type(2))) float        v2f;
typedef __attribute__((ext_vector_type(8))) float        v8f;
typedef __attribute__((ext_vector_type(4))) unsigned int u32x4;
typedef __attribute__((ext_vector_type(8))) int          i32x8;
typedef __attribute__((ext_vector_type(4))) int          i32x4;

#define B_DIM 8
#define C_DIM 128
#define L_DIM 4096
#define P_DIM 16
#define S_DIM 8
#define PC_DIM 511

// Branchless GELU: 0.5*h*(1+erf(h/sqrt(2))), erf via A&S 7.1.26 (|err|<=1.5e-7).
__device__ __forceinline__ float fast_gelu(float hv) {
  const float s   = hv * 0.70710678118654752f;
  const float ax  = fabsf(s);
  const float t   = __builtin_amdgcn_rcpf(fmaf(0.3275911f, ax, 1.0f));
  float p = fmaf(1.061405429f, t, -1.453152027f);
  p = fmaf(p, t, 1.421413741f);
  p = fmaf(p, t, -0.284496736f);
  p = fmaf(p, t, 0.254829592f);
  p = p * t;
  const float e       = __expf(-(ax * ax));
  const float erfc_ax = p * e;
  const float onep    = (s >= 0.0f) ? (2.0f - erfc_ax) : erfc_ax;  // 1+erf(s)
  return 0.5f * hv * onep;
}

// Sum over each 16-lane row via DPP16 xor-butterfly: masks {1,3,7,15} span the
// 16-lane subgroup and all map to DPP16 controls -> 4 VALU, no LDS traffic.
template <int CTRL>
__device__ __forceinline__ float dpp_addx(float v) {
  const int s = __builtin_amdgcn_update_dpp(0, __float_as_int(v),
                                            CTRL, 0xf, 0xf, true);
  return v + __int_as_float(s);
}
__device__ __forceinline__ float dpp_reduce16(float v) {
  v = dpp_addx<0x0B1>(v);  // quad_perm(1,0,3,2) : lane ^= 1
  v = dpp_addx<0x01B>(v);  // quad_perm(3,2,1,0) : lane ^= 3
  v = dpp_addx<0x141>(v);  // row_half_mirror    : lane ^= 7
  v = dpp_addx<0x140>(v);  // row_mirror         : lane ^= 15
  return v;
}

// Tensor Data Mover: 1-D row copy (4096 f32) global -> LDS.
// D# packing per CDNA5 ISA:
//  g0: [1:0]=count=1 | [63:32]=lds_addr | [120:64]=global_addr | [127:126]=type=2
//  g1: [17:16]=data_size=2 (4B) | tensor_dim0=4096 | tensor_dim1=1 |
//      tile_dim0=4096 | tile_dim1/2=0 | tensor_dim0_stride=4096
//  remaining groups: zero (tile is 1-D)
__device__ __forceinline__ void tdm_load_row(const float* gsrc,
                                             unsigned int lds_byte_addr) {
  const unsigned long long ga = (unsigned long long)(uintptr_t)gsrc;
  u32x4 g0 = { 1u,
               lds_byte_addr,
               (unsigned int)ga,
               (unsigned int)((ga >> 32) & 0x01FFFFFFu) | 0x80000000u };
  i32x8 g1 = { 0x20000,       // data_size = 4 bytes
               0x10000000,    // tensor_dim0 = 4096  (bits [79:48], low 16 here)
               0x00010000,    // tensor_dim1 = 1     (bits [111:80])
               0x10000000,    // tile_dim0  = 4096   (bits [127:112])
               0,             // tile_dim1 = tile_dim2 = 0 (1-D tile)
               4096,          // tensor_dim0_stride low 32
               0, 0 };
  i32x4 z4 = { 0, 0, 0, 0 };
  i32x8 z8 = { 0, 0, 0, 0, 0, 0, 0, 0 };
  __builtin_amdgcn_tensor_load_to_lds(g0, g1, z4, z4, z8, 0);
}

__global__ __launch_bounds__(256) void depatch_sampling_kernel(
    const float* __restrict__ x,  const float* __restrict__ w1,
    const float* __restrict__ b1, const float* __restrict__ w2,
    const float* __restrict__ b2, float* __restrict__ out)
{
  // Row c (patch source) padded to 4104 (masked garbage patch reads idx<=4103),
  // gather rows padded so x0+1 == 4096 reads 0.0 (reference validity mask).
  __shared__ float ldsP[4104];
  __shared__ float ldsA[4104];
  __shared__ float ldsB[4104];

  const int tid = threadIdx.x;
  const int bc  = blockIdx.x;          // b*128 + c
  const int b   = bc >> 7;
  const int c   = bc & 127;

  // ---- per-row-constant y coordinate chain (same float ops as reference) ----
  const float ys  = (float)c / 127.0f;
  const float gy  = (ys - 0.5f) * 2.0f;
  const float py  = (gy + 1.0f) * 0.5f * 127.0f;
  const float y0f = floorf(py);
  const float wy  = py - y0f;
  const float omwy = 1.0f - wy;
  const int   y0i = (int)y0f;
  const int   y1i = y0i + 1;
  const bool  vA  = (y0i >= 0) && (y0i < C_DIM);   // block-uniform
  const bool  vB  = (y1i >= 0) && (y1i < C_DIM);   // block-uniform

  const int lane = tid & 31;
  const int wave = tid >> 5;

  // ---- stage the three x rows into LDS with the Tensor Data Mover ----
  if (wave == 0) {
    tdm_load_row(x + (size_t)bc * L_DIM, (unsigned int)(uintptr_t)&ldsP[0]);
    if (vA) tdm_load_row(x + ((size_t)b * C_DIM + y0i) * L_DIM,
                         (unsigned int)(uintptr_t)&ldsA[0]);
    if (vB) tdm_load_row(x + ((size_t)b * C_DIM + y1i) * L_DIM,
                         (unsigned int)(uintptr_t)&ldsB[0]);
    __builtin_amdgcn_s_wait_tensorcnt(0);
  }
  if (!vA) {  // uniform: zero the out-of-range gather row
    const float4 z = make_float4(0.f, 0.f, 0.f, 0.f);
    float4* d4 = (float4*)ldsA;
    #pragma unroll
    for (int it = 0; it < 4; ++it) d4[it * 256 + tid] = z;
  }
  if (!vB) {
    const float4 z = make_float4(0.f, 0.f, 0.f, 0.f);
    float4* d4 = (float4*)ldsB;
    #pragma unroll
    for (int it = 0; it < 4; ++it) d4[it * 256 + tid] = z;
  }
  if (tid < 8) {
    ldsP[4096 + tid] = 0.0f;
    ldsA[4096 + tid] = 0.0f;
    ldsB[4096 + tid] = 0.0f;
  }
  __syncthreads();

  const int lh = lane & 15;   // N index within 16-tile / sample index p
  const int hi = lane >> 4;   // half-wave select

  // ---- preload B = w1^T (K=16 x N=64): 4 N-tiles x 4 K-steps ----
  // f32 WMMA 4x16 B layout: VGPR0 = (K=4s lanes0-15 | K=4s+2 lanes16-31),
  //                         VGPR1 = (K=4s+1         | K=4s+3), N = lane%16.
  v2f Bm[4][4];
  float b1l[4], w2l0[4], w2l1[4];
  #pragma unroll
  for (int n = 0; n < 4; ++n) {
    const int o = 16 * n + lh;
    b1l[n]  = b1[o];
    w2l0[n] = w2[o];
    w2l1[n] = w2[64 + o];
    #pragma unroll
    for (int s = 0; s < 4; ++s)
      Bm[n][s] = *(const v2f*)(w1 + o * 16 + 4 * s + 2 * hi);
  }
  const float b2_0 = b2[0];
  const float b2_1 = b2[1];
  const float tp   = (float)lh / 15.0f;   // t = p/(P-1)
  const float invL = 1.0f / 4095.0f;

  // ---- 32 tiles of 16 patches; wave w handles tiles w, w+8, w+16, w+24 ----
  #pragma unroll
  for (int it = 0; it < 4; ++it) {
    const int pc0 = (wave + 8 * it) * 16;

    const v8f vzero = {0.f,0.f,0.f,0.f,0.f,0.f,0.f,0.f};
    v8f acc[4] = {vzero, vzero, vzero, vzero};

    // GEMM1: h(16x64) = patches(16x16) x w1^T, K chained in 4 steps of 4
    #pragma unroll
    for (int s = 0; s < 4; ++s) {
      // A layout (f32 16x4): lane<16 holds (K=4s,4s+1) of row M=lane,
      //                      lane>=16 holds (K=4s+2,4s+3) of row M=lane-16.
      const v2f a = *(const v2f*)&ldsP[(pc0 + lh) * S_DIM + 4 * s + 2 * hi];
      #pragma unroll
      for (int n = 0; n < 4; ++n) {
        acc[n] = __builtin_amdgcn_wmma_f32_16x16x4_f32(
            false, a, false, Bm[n][s], (short)0, acc[n], false, false);
      }
    }

    // bias + GELU (branchless, EXEC stays all-ones through the WMMA region)
    #pragma unroll
    for (int n = 0; n < 4; ++n) {
      #pragma unroll
      for (int r = 0; r < 8; ++r) {
        acc[n][r] = fast_gelu(acc[n][r] + b1l[n]);
      }
    }

    // GEMM2 (N=2) + coordinate chain + bilinear sample
    #pragma unroll
    for (int r = 0; r < 8; ++r) {
      // lane holds h[M = r+8*hi, o = 16n+lh]; reduce over the 16-lane half
      float p0 = acc[0][r] * w2l0[0] + acc[1][r] * w2l0[1]
               + acc[2][r] * w2l0[2] + acc[3][r] * w2l0[3];
      float p1 = acc[0][r] * w2l1[0] + acc[1][r] * w2l1[1]
               + acc[2][r] * w2l1[2] + acc[3][r] * w2l1[3];
      p0 = dpp_reduce16(p0);
      p1 = dpp_reduce16(p1);

      const int pc = pc0 + r + 8 * hi;
      const float dx  = p0 + b2_0;
      const float dsv = fmaxf(p1 + b2_1 + 7.5f, 0.0f);        // relu(rel1+(P-1)/2)
      const float anchor = (float)pc * 8.0f + 7.5f;           // pc*S + 0.5*(P-1)
      float lo = (dx + anchor - dsv) * invL;
      lo = fminf(fmaxf(lo, 0.0f), 1.0f);
      float hv = (dx + anchor + dsv) * invL;
      hv = fminf(fmaxf(hv, 0.0f), 1.0f);
      const float xs  = lo + (hv - lo) * tp;
      const float px  = xs * 4095.0f;                          // == ((xs-.5)*2+1)*.5*(L-1)
      const float x0f = floorf(px);
      const float wx  = px - x0f;
      const float omwx = 1.0f - wx;
      const int   x0i = (int)x0f;                              // in [0, 4095]

      const float v00 = ldsA[x0i];
      const float v01 = ldsA[x0i + 1];
      const float v10 = ldsB[x0i];
      const float v11 = ldsB[x0i + 1];
      const float res = omwy * (omwx * v00 + wx * v01)
                      + wy   * (omwx * v10 + wx * v11);

      if (pc < PC_DIM)
        out[((size_t)bc * PC_DIM + pc) * P_DIM + lh] = res;
    }
  }
}

extern "C" void kernel_launch(void* const* d_in, const int* in_sizes, int n_in,
                              void* d_out, int out_size, void* d_ws, size_t ws_size,
                              hipStream_t stream) {
  const float* x  = (const float*)d_in[0];
  const float* w1 = (const float*)d_in[1];
  const float* b1 = (const float*)d_in[2];
  const float* w2 = (const float*)d_in[3];
  const float* b2 = (const float*)d_in[4];
  float* out = (float*)d_out;

  dim3 grid(B_DIM * C_DIM);   // 1024 workgroups: one per (b, c)
  dim3 block(256);            // 8 wave32s
  hipLaunchKernelGGL(depatch_sampling_kernel, grid, block, 0, stream,
                     x, w1, b1, w2, b2, out);
}